// HGPFlowModel_87686052315375
// MI455X (gfx1250) — compile-verified
//
#include <hip/hip_runtime.h>
#include <hip/hip_bf16.h>
#include <math.h>

typedef __attribute__((ext_vector_type(2))) float v2f;
typedef __attribute__((ext_vector_type(4))) float v4f;
typedef __attribute__((ext_vector_type(8))) float v8f;
typedef int gv4i __attribute__((vector_size(16)));   // matches builtin's V4i param

#define AS1 __attribute__((address_space(1)))
#define AS3 __attribute__((address_space(3)))

#define EPSV 1e-8f

constexpr int Bc = 16, Nc = 1024, Hc = 1280, Dc = 256;

// Output layout (floats), concatenated in reference return order
constexpr int OFF_CK  = 0;                       // charged_kin       (B,H,3)
constexpr int OFF_NK  = OFF_CK  + Bc * Hc * 3;   // neut_kin          (B,H,3)
constexpr int OFF_PIC = OFF_NK  + Bc * Hc * 3;   // proxy_is_charged  (B,H)
constexpr int OFF_EMF = OFF_PIC + Bc * Hc;       // proxy_em_frac     (B,H)
constexpr int OFF_INF = OFF_EMF + Bc * Hc;       // inc_times_node_f  (B,H,D)
constexpr int OFF_NFS = OFF_INF + Bc * Hc * Dc;  // node_feat_sum     (B,D)

#if __has_builtin(__builtin_amdgcn_global_load_async_to_lds_b128) && \
    __has_builtin(__builtin_amdgcn_global_load_async_to_lds_b32)
#define HAVE_ASYNC_LDS 1
#else
#define HAVE_ASYNC_LDS 0
#endif

#if __has_builtin(__builtin_amdgcn_s_wait_asynccnt)
#define WAIT_ASYNC(n) __builtin_amdgcn_s_wait_asynccnt(n)
#else
#define WAIT_ASYNC(n) asm volatile("s_wait_asynccnt %0" ::"i"(n) : "memory")
#endif

// ---------------------------------------------------------------------------
// Kernel 1: per-(b,n) precompute: wE = e_raw*is_topo, sin(phi), cos(phi)
// ---------------------------------------------------------------------------
__global__ __launch_bounds__(256) void prep_kernel(
    const float* __restrict__ e_raw, const float* __restrict__ phi,
    const int* __restrict__ is_topo,
    float* __restrict__ wE, float* __restrict__ sphi, float* __restrict__ cphi) {
  int i = blockIdx.x * blockDim.x + threadIdx.x;
  if (i < Bc * Nc) {
    wE[i] = e_raw[i] * (float)is_topo[i];
    float p = phi[i];
    sphi[i] = sinf(p);
    cphi[i] = cosf(p);
  }
}

// ---------------------------------------------------------------------------
// Kernel 2: one wave32 per (b,h) row: 5 weighted sums over N with weight
// w_n = pred_inc * e_raw * is_topo, then scalar epilogue.
// ---------------------------------------------------------------------------
__global__ __launch_bounds__(256) void row_reduce_kernel(
    const float* __restrict__ pred_inc, const float* __restrict__ wE,
    const float* __restrict__ eta_raw, const float* __restrict__ sphi,
    const float* __restrict__ cphi, const float* __restrict__ em_frac,
    const float* __restrict__ track_pt, const float* __restrict__ track_eta,
    const float* __restrict__ track_phi, const int* __restrict__ is_track,
    float* __restrict__ out) {
  int gwave = (int)((blockIdx.x * blockDim.x + threadIdx.x) >> 5);
  int lane  = threadIdx.x & 31;
  int b = gwave / Hc;
  int h = gwave % Hc;

  const float* prow = pred_inc + (size_t)(b * Hc + h) * Nc;
  const float* wEb  = wE      + (size_t)b * Nc;
  const float* etab = eta_raw + (size_t)b * Nc;
  const float* sb   = sphi    + (size_t)b * Nc;
  const float* cb   = cphi    + (size_t)b * Nc;
  const float* emb  = em_frac + (size_t)b * Nc;

  float S = 0.f, Te = 0.f, Ts = 0.f, Tc = 0.f, Tm = 0.f;
  for (int n = lane; n < Nc; n += 32) {
    float w = prow[n] * wEb[n];
    S += w;
    Te = fmaf(w, etab[n], Te);
    Ts = fmaf(w, sb[n],   Ts);
    Tc = fmaf(w, cb[n],   Tc);
    Tm = fmaf(w, emb[n],  Tm);
  }
  for (int off = 16; off > 0; off >>= 1) {
    S  += __shfl_xor(S,  off);
    Te += __shfl_xor(Te, off);
    Ts += __shfl_xor(Ts, off);
    Tc += __shfl_xor(Tc, off);
    Tm += __shfl_xor(Tm, off);
  }
  if (lane == 0) {
    bool charged = (h < Nc) && (is_track[b * Nc + h] != 0);
    float pt = 0.f, te = 0.f, tp = 0.f;
    if (charged) {
      pt = track_pt[b * Nc + h];
      te = track_eta[b * Nc + h];
      tp = track_phi[b * Nc + h];
    }
    int row = b * Hc + h;
    out[OFF_CK + row * 3 + 0] = pt;
    out[OFF_CK + row * 3 + 1] = te;
    out[OFF_CK + row * 3 + 2] = tp;

    float inv   = 1.0f / (S + EPSV);
    float nmask = charged ? 0.f : 1.f;
    out[OFF_NK + row * 3 + 0] = log1pf(fmaxf(S, 0.f)) * nmask;
    out[OFF_NK + row * 3 + 1] = (Te * inv) * (1.0f / 3.0f) * nmask;
    out[OFF_NK + row * 3 + 2] = atan2f(Ts * inv, Tc * inv) * nmask;

    out[OFF_PIC + row] = charged ? 1.f : 0.f;
    out[OFF_EMF + row] = Tm * inv;
  }
}

// ---------------------------------------------------------------------------
// Kernel 3: batched fp32 GEMM via V_WMMA_F32_16X16X4_F32
// C[b,h,d] = sum_n pred_inc[b,h,n] * node_feat[b,n,d]
// Block = 256 threads (8 waves). Block tile: 128 (H) x 64 (D). K chunk = 32.
// Each wave: one 16-row M tile, four 16-col N tiles (4 x v8f accumulators).
// B tile K-TRANSPOSED in LDS so every fragment pair is one aligned
// ds_load_b64 straight into WMMA fragment layout. LDS row stride 36 floats:
// rows 16B-aligned (b128 asyncs), stride-36 column reads bank-conflict-free.
// DOUBLE-BUFFERED async pipeline: issue tile i+1's GLOBAL_LOAD_ASYNC_TO_LDS
// into the other buffer, then wait ASYNCcnt <= 12 (tile i's 12 per-wave
// asyncs done; in-order completion). Trailing __syncthreads() (release fence)
// protects the buffer being overwritten next iteration.
// ---------------------------------------------------------------------------
#define TM 128
#define TN 64
#define KCH 32
#define LSTR 36
#define ASYNCS_PER_TILE 12   // 4 x b128 (A) + 8 x b32 (B) per thread

__global__ __launch_bounds__(256) void gemm_wmma_kernel(
    const float* __restrict__ pred_inc, const float* __restrict__ node_feat,
    float* __restrict__ out) {
  __shared__ __align__(16) float As[2][TM][LSTR];   // 128 x 32 used
  __shared__ __align__(16) float BsT[2][TN][LSTR];  // [d][k], 64 x 32 used

  int b  = blockIdx.z;
  int h0 = blockIdx.y * TM;
  int d0 = blockIdx.x * TN;
  int tid  = threadIdx.x;
  int wave = tid >> 5;
  int lane = tid & 31;

  const float* Abase = pred_inc  + (size_t)b * Hc * Nc;
  const float* Bbase = node_feat + (size_t)b * Nc * Dc;

  v8f acc0 = {}, acc1 = {}, acc2 = {}, acc3 = {};

  // Fragment lane mapping (fp32 16x4 A / 4x16 B): lanes 0-15 -> K={0,1},
  // lanes 16-31 -> K={2,3}; M/N index = lane & 15.
  int fm   = lane & 15;
  int fkb  = (lane >> 4) * 2;
  int mrow = wave * 16 + fm;

  // Cooperative-load indexing
  int a_c4  = (tid & 7) * 4;   // A: 8 lanes x float4 cover a 32-float row
  int a_r   = tid >> 3;        // A: 32 rows per pass, 4 passes
  int b_d   = tid & 63;        // B: 64 consecutive d lanes (coalesced)
  int b_k   = tid >> 6;        // B: 4 k rows per pass, 8 passes

#if HAVE_ASYNC_LDS
  auto issue_tile = [&](int kbase, int bsel) {
    // A tile: 128x32, 16B-contiguous chunks, async b128 -> LDS
    for (int p = 0; p < 4; ++p) {
      int r = a_r + p * 32;
      __builtin_amdgcn_global_load_async_to_lds_b128(
          (AS1 gv4i*)&Abase[(size_t)(h0 + r) * Nc + kbase + a_c4],
          (AS3 gv4i*)&As[bsel][r][a_c4], 0, 0);
    }
    // B tile: 32x64 read coalesced over d, scattered transposed into LDS
    for (int p = 0; p < 8; ++p) {
      int k = b_k + p * 4;
      __builtin_amdgcn_global_load_async_to_lds_b32(
          (AS1 int*)&Bbase[(size_t)(kbase + k) * Dc + d0 + b_d],
          (AS3 int*)&BsT[bsel][b_d][k], 0, 0);
    }
  };

  issue_tile(0, 0);
  int buf = 0;
  for (int k0 = 0; k0 < Nc; k0 += KCH) {
    if (k0 + KCH < Nc) {
      issue_tile(k0 + KCH, buf ^ 1);
      WAIT_ASYNC(ASYNCS_PER_TILE);   // tile i done; tile i+1 may be in flight
    } else {
      WAIT_ASYNC(0);
    }
    __syncthreads();

#pragma unroll
    for (int kk = 0; kk < KCH; kk += 4) {
      v2f a   = *(const v2f*)&As[buf][mrow][kk + fkb];
      v2f bf0 = *(const v2f*)&BsT[buf][fm +  0][kk + fkb];
      v2f bf1 = *(const v2f*)&BsT[buf][fm + 16][kk + fkb];
      v2f bf2 = *(const v2f*)&BsT[buf][fm + 32][kk + fkb];
      v2f bf3 = *(const v2f*)&BsT[buf][fm + 48][kk + fkb];
      acc0 = __builtin_amdgcn_wmma_f32_16x16x4_f32(false, a, false, bf0,
                                                   (short)0, acc0, false, false);
      acc1 = __builtin_amdgcn_wmma_f32_16x16x4_f32(false, a, false, bf1,
                                                   (short)0, acc1, false, false);
      acc2 = __builtin_amdgcn_wmma_f32_16x16x4_f32(false, a, false, bf2,
                                                   (short)0, acc2, false, false);
      acc3 = __builtin_amdgcn_wmma_f32_16x16x4_f32(false, a, false, bf3,
                                                   (short)0, acc3, false, false);
    }
    __syncthreads();   // all waves done reading buf before it is re-filled
    buf ^= 1;
  }
#else
  for (int k0 = 0; k0 < Nc; k0 += KCH) {
    for (int p = 0; p < 4; ++p) {
      int r = a_r + p * 32;
      *(v4f*)&As[0][r][a_c4] =
          *(const v4f*)&Abase[(size_t)(h0 + r) * Nc + k0 + a_c4];
    }
    for (int p = 0; p < 8; ++p) {
      int k = b_k + p * 4;
      BsT[0][b_d][k] = Bbase[(size_t)(k0 + k) * Dc + d0 + b_d];
    }
    __syncthreads();
    if (k0 + KCH < Nc) {
      __builtin_prefetch(&Abase[(size_t)(h0 + a_r) * Nc + k0 + KCH + a_c4], 0, 0);
      __builtin_prefetch(&Bbase[(size_t)(k0 + KCH + b_k) * Dc + d0 + b_d], 0, 0);
    }
#pragma unroll
    for (int kk = 0; kk < KCH; kk += 4) {
      v2f a   = *(const v2f*)&As[0][mrow][kk + fkb];
      v2f bf0 = *(const v2f*)&BsT[0][fm +  0][kk + fkb];
      v2f bf1 = *(const v2f*)&BsT[0][fm + 16][kk + fkb];
      v2f bf2 = *(const v2f*)&BsT[0][fm + 32][kk + fkb];
      v2f bf3 = *(const v2f*)&BsT[0][fm + 48][kk + fkb];
      acc0 = __builtin_amdgcn_wmma_f32_16x16x4_f32(false, a, false, bf0,
                                                   (short)0, acc0, false, false);
      acc1 = __builtin_amdgcn_wmma_f32_16x16x4_f32(false, a, false, bf1,
                                                   (short)0, acc1, false, false);
      acc2 = __builtin_amdgcn_wmma_f32_16x16x4_f32(false, a, false, bf2,
                                                   (short)0, acc2, false, false);
      acc3 = __builtin_amdgcn_wmma_f32_16x16x4_f32(false, a, false, bf3,
                                                   (short)0, acc3, false, false);
    }
    __syncthreads();
  }
#endif

  // C/D 16x16 fp32 layout: VGPR r -> M = r (lanes 0-15) / r+8 (lanes 16-31)
  int rbase = h0 + wave * 16 + ((lane >> 4) * 8);
  int col   = d0 + (lane & 15);
  float* obase = out + OFF_INF + (size_t)b * Hc * Dc;
#pragma unroll
  for (int r = 0; r < 8; ++r) {
    float* orow = obase + (size_t)(rbase + r) * Dc + col;
    orow[0]  = acc0[r];
    orow[16] = acc1[r];
    orow[32] = acc2[r];
    orow[48] = acc3[r];
  }
}

// ---------------------------------------------------------------------------
// Kernel 4: node_feat_sum[b,d] = sum_n node_feat[b,n,d] (coalesced over d)
// ---------------------------------------------------------------------------
__global__ __launch_bounds__(256) void nfsum_kernel(
    const float* __restrict__ node_feat, float* __restrict__ out) {
  int t = blockIdx.x * blockDim.x + threadIdx.x;  // 0..B*D-1
  int b = t / Dc;
  int d = t % Dc;
  const float* base = node_feat + (size_t)b * Nc * Dc + d;
  float s = 0.f;
  for (int n = 0; n < Nc; ++n) s += base[(size_t)n * Dc];
  out[OFF_NFS + t] = s;
}

// ---------------------------------------------------------------------------
extern "C" void kernel_launch(void* const* d_in, const int* in_sizes, int n_in,
                              void* d_out, int out_size, void* d_ws, size_t ws_size,
                              hipStream_t stream) {
  (void)in_sizes; (void)n_in; (void)out_size; (void)ws_size;
  const float* pred_inc  = (const float*)d_in[0];
  const float* node_feat = (const float*)d_in[1];
  const float* e_raw     = (const float*)d_in[2];
  const float* eta_raw   = (const float*)d_in[3];
  const float* phi       = (const float*)d_in[4];
  const float* em_frac   = (const float*)d_in[5];
  const float* track_pt  = (const float*)d_in[6];
  const float* track_eta = (const float*)d_in[7];
  const float* track_phi = (const float*)d_in[8];
  const int*   is_track  = (const int*)d_in[9];
  const int*   is_topo   = (const int*)d_in[10];

  float* out  = (float*)d_out;
  float* ws   = (float*)d_ws;
  float* wE   = ws;
  float* sphi = ws + Bc * Nc;
  float* cphi = ws + 2 * Bc * Nc;

  prep_kernel<<<(Bc * Nc) / 256, 256, 0, stream>>>(e_raw, phi, is_topo, wE, sphi, cphi);

  // one wave per (b,h): B*H = 20480 waves, 8 waves/block
  row_reduce_kernel<<<(Bc * Hc) / 8, 256, 0, stream>>>(
      pred_inc, wE, eta_raw, sphi, cphi, em_frac,
      track_pt, track_eta, track_phi, is_track, out);

  dim3 ggrid(Dc / TN, Hc / TM, Bc);  // (4, 10, 16)
  gemm_wmma_kernel<<<ggrid, 256, 0, stream>>>(pred_inc, node_feat, out);

  nfsum_kernel<<<(Bc * Dc) / 256, 256, 0, stream>>>(node_feat, out);
}